// HungarianMatcher_7387343749451
// MI455X (gfx1250) — compile-verified
//
#include <hip/hip_runtime.h>
#include <hip/hip_bf16.h>

#define ALPHA   0.25f
#define W_CLASS 2.0f
#define W_BBOX  5.0f
#define W_GIOU  2.0f
#define EPS_F   1e-8f

#define BS  16
#define QN  900
#define NC  80
#define NQ  (BS * QN)   // 14400
#define TT  1600        // total targets

typedef __attribute__((ext_vector_type(2))) float v2f;
typedef __attribute__((ext_vector_type(8))) float v8f;

// ---------------------------------------------------------------------------
// Kernel 1: cls_diff[q*80+c] = pos_cost - neg_cost  (focal-style class cost)
// ---------------------------------------------------------------------------
__global__ __launch_bounds__(256) void cls_cost_kernel(
    const float* __restrict__ logits, float* __restrict__ cls_diff)
{
    int idx = blockIdx.x * blockDim.x + threadIdx.x;
    if (idx >= NQ * NC) return;
    float x = logits[idx];
    float p = 1.0f / (1.0f + expf(-x));
    float neg = (1.0f - ALPHA) * p * p * (-logf(1.0f - p + EPS_F));
    float pos = ALPHA * (1.0f - p) * (1.0f - p) * (-logf(p + EPS_F));
    cls_diff[idx] = pos - neg;
}

// ---------------------------------------------------------------------------
// Kernel 2: tiled 64x64 cost matrix.
//   cost_class via one-hot WMMA gather (exact in f32),
//   L1 + GIoU elementwise from LDS-staged boxes.
// ---------------------------------------------------------------------------
__global__ __launch_bounds__(256) void matcher_main(
    const float* __restrict__ pred_boxes,   // [NQ,4] cxcywh
    const float* __restrict__ tgt_boxes,    // [TT,4] xyxy
    const int*   __restrict__ tgt_labels,   // [TT]
    const float* __restrict__ cls_diff,     // [NQ,NC]
    float* __restrict__ out)                // [NQ,TT]
{
    __shared__ float qc[64][4];  // query cxcywh
    __shared__ float qx[64][4];  // query xyxy
    __shared__ float qa[64];     // query area
    __shared__ float tc[64][4];  // target cxcywh
    __shared__ float txy[64][4]; // target xyxy
    __shared__ float ta[64];     // target area
    __shared__ int   tl[64];     // target labels

    const int tid  = threadIdx.x;
    const int qblk = blockIdx.x * 64;
    const int tblk = blockIdx.y * 64;

    if (tid < 64) {
        const float4 b = ((const float4*)pred_boxes)[qblk + tid];
        qc[tid][0] = b.x; qc[tid][1] = b.y; qc[tid][2] = b.z; qc[tid][3] = b.w;
        float x1 = b.x - 0.5f * b.z, y1 = b.y - 0.5f * b.w;
        float x2 = b.x + 0.5f * b.z, y2 = b.y + 0.5f * b.w;
        qx[tid][0] = x1; qx[tid][1] = y1; qx[tid][2] = x2; qx[tid][3] = y2;
        qa[tid] = (x2 - x1) * (y2 - y1);
    } else if (tid < 128) {
        const int j = tid - 64;
        const float4 b = ((const float4*)tgt_boxes)[tblk + j];
        txy[j][0] = b.x; txy[j][1] = b.y; txy[j][2] = b.z; txy[j][3] = b.w;
        tc[j][0] = (b.x + b.z) * 0.5f;
        tc[j][1] = (b.y + b.w) * 0.5f;
        tc[j][2] = b.z - b.x;
        tc[j][3] = b.w - b.y;
        ta[j] = (b.z - b.x) * (b.w - b.y);
        tl[j] = tgt_labels[tblk + j];
    }
    __syncthreads();

    const int wave  = tid >> 5;
    const int lane  = tid & 31;
    const int wm    = wave & 3;    // M sub-tile (0..3), 16 rows each
    const int wn    = wave >> 1 >> 1; // wave/4: N group (0..1), 32 cols each
    const int lhalf = lane >> 4;   // lane half (0/1)
    const int lmod  = lane & 15;

    // A fragment source row: M = lane%16 (per ISA 32-bit A 16x4 layout),
    // K = vgpr + 2*(lane/16) within each k-step of 4.
    const float* arow = cls_diff + (size_t)(qblk + wm * 16 + lmod) * NC;

    // B one-hot columns for the two 16-wide N tiles this wave owns.
    const int nl0  = wn * 32 + lmod;       // N = lane%16 (per B layout)
    const int nl1  = nl0 + 16;
    const int lab0 = tl[nl0];
    const int lab1 = tl[nl1];

    v8f c0 = {0.f,0.f,0.f,0.f,0.f,0.f,0.f,0.f};
    v8f c1 = {0.f,0.f,0.f,0.f,0.f,0.f,0.f,0.f};

#pragma unroll
    for (int k0 = 0; k0 < NC / 4; ++k0) {
        const int kbase = k0 * 4 + 2 * lhalf;        // K of vgpr0; vgpr1 = K+1
        v2f a = *(const v2f*)(arow + kbase);          // global_load_b64
        v2f b0 = { (lab0 == kbase) ? 1.0f : 0.0f,
                   (lab0 == kbase + 1) ? 1.0f : 0.0f };
        v2f b1 = { (lab1 == kbase) ? 1.0f : 0.0f,
                   (lab1 == kbase + 1) ? 1.0f : 0.0f };
        c0 = __builtin_amdgcn_wmma_f32_16x16x4_f32(false, a, false, b0,
                                                   (short)0, c0, false, false);
        c1 = __builtin_amdgcn_wmma_f32_16x16x4_f32(false, a, false, b1,
                                                   (short)0, c1, false, false);
    }

    // Elementwise L1 + GIoU and final store.
    // C/D layout: element (m, n) with m = v + 8*(lane/16), n = lane%16.
#pragma unroll
    for (int v = 0; v < 8; ++v) {
        const int m  = v + 8 * lhalf;
        const int ql = wm * 16 + m;
        const float pcx = qc[ql][0], pcy = qc[ql][1];
        const float pw  = qc[ql][2], ph  = qc[ql][3];
        const float ax1 = qx[ql][0], ay1 = qx[ql][1];
        const float ax2 = qx[ql][2], ay2 = qx[ql][3];
        const float areaA = qa[ql];
        const size_t orow = (size_t)(qblk + ql) * TT + tblk;
#pragma unroll
        for (int t2 = 0; t2 < 2; ++t2) {
            const int nloc = wn * 32 + t2 * 16 + lmod;
            // L1 in cxcywh
            float l1 = fabsf(pcx - tc[nloc][0]) + fabsf(pcy - tc[nloc][1])
                     + fabsf(pw  - tc[nloc][2]) + fabsf(ph  - tc[nloc][3]);
            // GIoU in xyxy
            const float bx1 = txy[nloc][0], by1 = txy[nloc][1];
            const float bx2 = txy[nloc][2], by2 = txy[nloc][3];
            const float ltx = fmaxf(ax1, bx1), lty = fmaxf(ay1, by1);
            const float rbx = fminf(ax2, bx2), rby = fminf(ay2, by2);
            const float iw = fmaxf(rbx - ltx, 0.0f);
            const float ih = fmaxf(rby - lty, 0.0f);
            const float inter = iw * ih;
            const float uni   = areaA + ta[nloc] - inter;
            const float iou   = inter / uni;
            const float cx1 = fminf(ax1, bx1), cy1 = fminf(ay1, by1);
            const float cx2 = fmaxf(ax2, bx2), cy2 = fmaxf(ay2, by2);
            const float cw = fmaxf(cx2 - cx1, 0.0f);
            const float ch = fmaxf(cy2 - cy1, 0.0f);
            const float areaC = cw * ch;
            const float giou  = iou - (areaC - uni) / areaC;
            const float cls   = (t2 == 0) ? c0[v] : c1[v];
            out[orow + nloc] = W_BBOX * l1 + W_CLASS * cls + W_GIOU * (-giou);
        }
    }
}

// ---------------------------------------------------------------------------
extern "C" void kernel_launch(void* const* d_in, const int* in_sizes, int n_in,
                              void* d_out, int out_size, void* d_ws, size_t ws_size,
                              hipStream_t stream)
{
    const float* pred_logits = (const float*)d_in[0];  // [16,900,80]
    const float* pred_boxes  = (const float*)d_in[1];  // [16,900,4] cxcywh
    const float* tgt_boxes   = (const float*)d_in[2];  // [1600,4] xyxy
    const int*   tgt_labels  = (const int*)d_in[3];    // [1600]
    float* out      = (float*)d_out;                   // [14400,1600]
    float* cls_diff = (float*)d_ws;                    // [14400,80] = 4.6 MB

    const int n1 = NQ * NC;
    cls_cost_kernel<<<(n1 + 255) / 256, 256, 0, stream>>>(pred_logits, cls_diff);

    dim3 grid(NQ / 64, TT / 64);   // (225, 25)
    matcher_main<<<grid, 256, 0, stream>>>(pred_boxes, tgt_boxes, tgt_labels,
                                           cls_diff, out);
}